// LocalQKConv_58557584114036
// MI455X (gfx1250) — compile-verified
//
#include <hip/hip_runtime.h>

// Problem constants (match reference)
#define B_  8
#define N_  8192
#define H_  128
#define G_  16
#define MH_ 16
#define EPS_ 1e-8f

typedef __attribute__((ext_vector_type(2))) float v2f;
typedef __attribute__((ext_vector_type(8))) float v8f;

__device__ __forceinline__ float sigmoidf_(float x) { return 1.0f / (1.0f + __expf(-x)); }

// ---------------------------------------------------------------------------
// Kernel 1: vec_geom[b,n,c,g] = sum_h vec[b,n,c,h] * W_geom[h,g]
// M = B*N*3 rows, K = 128, Ncols = 16.  One 16x16 tile per wave via chained
// V_WMMA_F32_16X16X4_F32 (full fp32 precision, 32 K-steps).
// ---------------------------------------------------------------------------
__global__ void __launch_bounds__(256) vecgeom_wmma(const float* __restrict__ vec,
                                                    const float* __restrict__ Wg,
                                                    float* __restrict__ vg) {
    __shared__ __align__(16) float sW[H_ * G_];   // 128x16 = 8KB
    for (int t = threadIdx.x; t < H_ * G_; t += 256) sW[t] = Wg[t];
    __syncthreads();

    const int wave = threadIdx.x >> 5;
    const int lane = threadIdx.x & 31;
    const int m    = lane & 15;          // row within tile / col within N
    const int kh   = (lane >> 4) * 2;    // K sub-offset per half-wave
    const int rowBase = (blockIdx.x * 8 + wave) * 16;

    const float* arow = vec + (size_t)(rowBase + m) * H_;
    v8f acc = {};
#pragma unroll
    for (int kk = 0; kk < 32; ++kk) {
        const int k0 = kk * 4 + kh;
        v2f a = *(const v2f*)(arow + k0);             // A[m][k0], A[m][k0+1]
        v2f b;
        b.x = sW[k0 * G_ + m];                        // B[k0][m]
        b.y = sW[(k0 + 1) * G_ + m];                  // B[k0+1][m]
        acc = __builtin_amdgcn_wmma_f32_16x16x4_f32(false, a, false, b,
                                                    (short)0, acc, false, false);
    }
    const int rhi = (lane >> 4) * 8;
#pragma unroll
    for (int r = 0; r < 8; ++r) {
        const int grow = rowBase + r + rhi;           // C layout: M = r (+8 hi half)
        vg[(size_t)grow * G_ + m] = acc[r];
    }
}

// ---------------------------------------------------------------------------
// Kernel 2: u[b,i,:,g] = sum over valid offsets of unit(vec_geom[j]-vec_geom[i])
// One thread per (b, n, g); g-contiguous => coalesced.
// ---------------------------------------------------------------------------
__global__ void __launch_bounds__(256) build_u(const float* __restrict__ vg,
                                               float* __restrict__ u) {
    const int t = blockIdx.x * 256 + threadIdx.x;     // < B*N*G = 1048576
    const int g = t & (G_ - 1);
    const int n = (t >> 4) & (N_ - 1);
    const int b = t >> 17;                            // t / (N_*G_)
    const size_t bi = ((size_t)(b * N_ + n) * 3) * G_ + g;
    const float vi0 = vg[bi], vi1 = vg[bi + G_], vi2 = vg[bi + 2 * G_];
    float u0 = 0.f, u1 = 0.f, u2 = 0.f;
#pragma unroll
    for (int oi = 0; oi < 6; ++oi) {
        const int off = (oi < 3) ? (oi - 3) : (oi - 2);   // {-3,-2,-1,1,2,3}
        const int j = n + off;
        if (j < 0 || j >= N_) continue;
        const size_t bj = ((size_t)(b * N_ + j) * 3) * G_ + g;
        const float b0 = vg[bj] - vi0;
        const float b1 = vg[bj + G_] - vi1;
        const float b2 = vg[bj + 2 * G_] - vi2;
        const float inv = 1.0f / fmaxf(sqrtf(b0 * b0 + b1 * b1 + b2 * b2), EPS_);
        u0 += b0 * inv; u1 += b1 * inv; u2 += b2 * inv;
    }
    u[bi] = u0; u[bi + G_] = u1; u[bi + 2 * G_] = u2;
}

// ---------------------------------------------------------------------------
// Kernel 3: per 16-node tile: edge feats + MLP -> g_geom (96x16 each for q,k),
// then fp32 WMMA (96x16 @ 16x128), sigmoid, gate x[dst], scatter-add to q/k.
// ---------------------------------------------------------------------------
__global__ void __launch_bounds__(256) edge_gate_wmma(
    const float* __restrict__ x,   const float* __restrict__ vg,
    const float* __restrict__ u,
    const float* __restrict__ qw1, const float* __restrict__ qb1,
    const float* __restrict__ qw2, const float* __restrict__ qb2,
    const float* __restrict__ kw1, const float* __restrict__ kb1,
    const float* __restrict__ kw2, const float* __restrict__ kb2,
    const float* __restrict__ Wq,  const float* __restrict__ bq,
    const float* __restrict__ Wk,  const float* __restrict__ bk,
    float* __restrict__ qout, float* __restrict__ kout) {
    __shared__ __align__(16) float sWq[G_ * H_];     // 8KB
    __shared__ __align__(16) float sWk[G_ * H_];     // 8KB
    __shared__ float sbq[H_], sbk[H_];
    __shared__ float s_qw1[2 * MH_], s_qb1[MH_], s_qw2[MH_];
    __shared__ float s_kw1[2 * MH_], s_kb1[MH_], s_kw2[MH_];
    __shared__ float s_qb2, s_kb2;
    __shared__ __align__(16) float sgq[96 * G_];     // 6KB
    __shared__ __align__(16) float sgk[96 * G_];     // 6KB
    __shared__ float sqa[16 * H_];                   // 8KB
    __shared__ float ska[16 * H_];                   // 8KB

    const int tid = threadIdx.x;
    const int b   = blockIdx.x / (N_ / 16);
    const int i0  = (blockIdx.x % (N_ / 16)) * 16;

    for (int t = tid; t < G_ * H_; t += 256) { sWq[t] = Wq[t]; sWk[t] = Wk[t]; }
    for (int t = tid; t < H_; t += 256)      { sbq[t] = bq[t]; sbk[t] = bk[t]; }
    if (tid < 2 * MH_) { s_qw1[tid] = qw1[tid]; s_kw1[tid] = kw1[tid]; }
    if (tid < MH_) {
        s_qb1[tid] = qb1[tid]; s_qw2[tid] = qw2[tid];
        s_kb1[tid] = kb1[tid]; s_kw2[tid] = kw2[tid];
    }
    if (tid == 0) { s_qb2 = qb2[0]; s_kb2 = kb2[0]; }
    for (int t = tid; t < 16 * H_; t += 256) { sqa[t] = 0.f; ska[t] = 0.f; }
    __syncthreads();

    // ---- Phase 1: edge features + MLP -> g_geom (one item per (edge,g)) ----
    for (int it = 0; it < 6; ++it) {
        const int item = tid + it * 256;          // 0..1535
        const int g    = item & 15;
        const int e    = item >> 4;               // 0..95
        const int node = e / 6, oi = e % 6;
        const int off  = (oi < 3) ? (oi - 3) : (oi - 2);
        const int i    = i0 + node;
        const int j    = i + off;
        const int jc   = min(max(j, 0), N_ - 1);  // clamp for safe load
        const size_t bi = ((size_t)(b * N_ + i) * 3) * G_ + g;
        const size_t bj = ((size_t)(b * N_ + jc) * 3) * G_ + g;

        const float d0 = vg[bj] - vg[bi];
        const float d1 = vg[bj + G_] - vg[bi + G_];
        const float d2 = vg[bj + 2 * G_] - vg[bi + 2 * G_];
        const float inv = 1.0f / fmaxf(sqrtf(d0 * d0 + d1 * d1 + d2 * d2), EPS_);
        const float e0 = d0 * inv, e1 = d1 * inv, e2 = d2 * inv;

        const float ui0 = u[bi], ui1 = u[bi + G_], ui2 = u[bi + 2 * G_];
        const float uj0 = u[bj], uj1 = u[bj + G_], uj2 = u[bj + 2 * G_];

        const float dui = ui0 * e0 + ui1 * e1 + ui2 * e2;
        const float nui = fmaxf(sqrtf(ui0 * ui0 + ui1 * ui1 + ui2 * ui2), EPS_);
        const float ang = fminf(fmaxf(dui / nui, -1.f), 1.f);

        const float p0 = ui0 - dui * e0, p1 = ui1 - dui * e1, p2 = ui2 - dui * e2;
        const float duj = uj0 * e0 + uj1 * e1 + uj2 * e2;
        const float w0 = uj0 - duj * e0, w1 = uj1 - duj * e1, w2 = uj2 - duj * e2;
        const float dpp = p0 * w0 + p1 * w1 + p2 * w2;
        const float npi = fmaxf(sqrtf(p0 * p0 + p1 * p1 + p2 * p2), EPS_);
        const float npj = fmaxf(sqrtf(w0 * w0 + w1 * w1 + w2 * w2), EPS_);
        const float dih = fminf(fmaxf(dpp / fmaxf(npi * npj, EPS_), -1.f), 1.f);

        float accq = s_qb2, acck = s_kb2;
#pragma unroll
        for (int mh = 0; mh < MH_; ++mh) {
            const float hq = ang * s_qw1[mh] + dih * s_qw1[MH_ + mh] + s_qb1[mh];
            accq += (hq * sigmoidf_(hq)) * s_qw2[mh];
            const float hk = ang * s_kw1[mh] + dih * s_kw1[MH_ + mh] + s_kb1[mh];
            acck += (hk * sigmoidf_(hk)) * s_kw2[mh];
        }
        sgq[e * G_ + g] = accq;
        sgk[e * G_ + g] = acck;
    }
    __syncthreads();

    // ---- Phase 2: WMMA  (96x16) @ (16x128)  for q and k gates ----
    const int wave = tid >> 5;       // wave owns H-tile nt = wave
    const int lane = tid & 31;
    const int nt   = wave;
    const int m16  = lane & 15;
    const int kh   = (lane >> 4) * 2;
    const int rhi  = (lane >> 4) * 8;
    const int col  = nt * 16 + m16;

    for (int mt = 0; mt < 6; ++mt) {
        v8f aq = {}, ak = {};
#pragma unroll
        for (int kk = 0; kk < 4; ++kk) {
            const int k0  = kk * 4 + kh;
            const int row = mt * 16 + m16;
            v2f a1 = *(const v2f*)&sgq[row * G_ + k0];
            v2f a2 = *(const v2f*)&sgk[row * G_ + k0];
            v2f b1, b2;
            b1.x = sWq[k0 * H_ + col];  b1.y = sWq[(k0 + 1) * H_ + col];
            b2.x = sWk[k0 * H_ + col];  b2.y = sWk[(k0 + 1) * H_ + col];
            aq = __builtin_amdgcn_wmma_f32_16x16x4_f32(false, a1, false, b1,
                                                       (short)0, aq, false, false);
            ak = __builtin_amdgcn_wmma_f32_16x16x4_f32(false, a2, false, b2,
                                                       (short)0, ak, false, false);
        }
#pragma unroll
        for (int r = 0; r < 8; ++r) {
            const int e    = mt * 16 + r + rhi;       // edge row 0..95
            const int node = e / 6, oi = e % 6;
            const int off  = (oi < 3) ? (oi - 3) : (oi - 2);
            const int i    = i0 + node;
            const int j    = i + off;
            if (j >= 0 && j < N_) {
                const float sj = x[((size_t)(b * N_ + j)) * H_ + col];
                atomicAdd(&sqa[node * H_ + col], sigmoidf_(aq[r] + sbq[col]) * sj);
                atomicAdd(&ska[node * H_ + col], sigmoidf_(ak[r] + sbk[col]) * sj);
            }
        }
    }
    __syncthreads();

    // ---- Phase 3: write q,k tiles ----
    for (int t = tid; t < 16 * H_; t += 256) {
        const int node = t >> 7;
        const int h    = t & (H_ - 1);
        const size_t o = ((size_t)(b * N_ + i0 + node)) * H_ + h;
        qout[o] = sqa[t];
        kout[o] = ska[t];
    }
}

// ---------------------------------------------------------------------------
extern "C" void kernel_launch(void* const* d_in, const int* in_sizes, int n_in,
                              void* d_out, int out_size, void* d_ws, size_t ws_size,
                              hipStream_t stream) {
    (void)in_sizes; (void)n_in; (void)out_size; (void)ws_size;
    const float* x   = (const float*)d_in[0];
    const float* vec = (const float*)d_in[1];
    const float* Wg  = (const float*)d_in[2];
    const float* qw1 = (const float*)d_in[3];
    const float* qb1 = (const float*)d_in[4];
    const float* qw2 = (const float*)d_in[5];
    const float* qb2 = (const float*)d_in[6];
    const float* kw1 = (const float*)d_in[7];
    const float* kb1 = (const float*)d_in[8];
    const float* kw2 = (const float*)d_in[9];
    const float* kb2 = (const float*)d_in[10];
    const float* Wq  = (const float*)d_in[11];
    const float* bq  = (const float*)d_in[12];
    const float* Wk  = (const float*)d_in[13];
    const float* bk  = (const float*)d_in[14];

    float* qout = (float*)d_out;
    float* kout = qout + (size_t)B_ * N_ * H_;
    float* vg   = (float*)d_ws;                           // B*N*3*G floats
    float* u    = vg + (size_t)B_ * N_ * 3 * G_;          // B*N*3*G floats

    // K1: 196608 rows / (8 waves * 16 rows) = 1536 blocks
    vecgeom_wmma<<<(B_ * N_ * 3) / 128, 256, 0, stream>>>(vec, Wg, vg);
    // K2: B*N*G threads
    build_u<<<(B_ * N_ * G_) / 256, 256, 0, stream>>>(vg, u);
    // K3: one block per 16-node tile per batch row
    edge_gate_wmma<<<B_ * (N_ / 16), 256, 0, stream>>>(
        x, vg, u, qw1, qb1, qw2, qb2, kw1, kb1, kw2, kb2, Wq, bq, Wk, bk,
        qout, kout);
}